// GraphConvLayer_19052474925412
// MI455X (gfx1250) — compile-verified
//
#include <hip/hip_runtime.h>
#include <hip/hip_bf16.h>

// Problem constants from the reference
#define BB   256   // batch
#define NN   2048  // nodes
#define UU   128   // node units
#define GG   256   // graph units

typedef __attribute__((ext_vector_type(2))) float v2f;
typedef __attribute__((ext_vector_type(8))) float v8f;

// ---------------------------------------------------------------------------
// Kernel 1a: partial column sums of adj (N x N, row-major), float4-wide.
// grid (2, 16), block 256. Each thread owns 4 consecutive columns (one float4),
// each block.x covers 1024 columns, block.y a 128-row chunk.
// Wave reads 512 contiguous bytes per row per instruction (global_load_b128).
// ---------------------------------------------------------------------------
__global__ void colsum_partial_kernel(const float4* __restrict__ adj4,
                                      float4* __restrict__ partial4) {
    int j4    = blockIdx.x * 256 + threadIdx.x;   // float4 column-group (0..511)
    int chunk = blockIdx.y;
    int i0    = chunk * 128;
    float4 s = make_float4(0.f, 0.f, 0.f, 0.f);
    #pragma unroll 8
    for (int i = 0; i < 128; ++i) {
        float4 v = adj4[(size_t)(i0 + i) * (NN / 4) + j4];
        s.x += v.x; s.y += v.y; s.z += v.z; s.w += v.w;
    }
    partial4[chunk * (NN / 4) + j4] = s;
}

// ---------------------------------------------------------------------------
// Kernel 1b: reduce 16 partials -> c[j] = colsum / N, float4-wide.
// grid 2, block 256. Fixed-order reduction => deterministic.
// ---------------------------------------------------------------------------
__global__ void colsum_reduce_kernel(const float4* __restrict__ partial4,
                                     float4* __restrict__ c4) {
    int j4 = blockIdx.x * 256 + threadIdx.x;      // 0..511
    float4 s = make_float4(0.f, 0.f, 0.f, 0.f);
    #pragma unroll
    for (int t = 0; t < 16; ++t) {
        float4 v = partial4[t * (NN / 4) + j4];
        s.x += v.x; s.y += v.y; s.z += v.z; s.w += v.w;
    }
    const float inv = 1.0f / (float)NN;
    s.x *= inv; s.y *= inv; s.z *= inv; s.w *= inv;
    c4[j4] = s;
}

// ---------------------------------------------------------------------------
// Kernel 2: x_mean[b,u] = sum_j c[j] * relu(f[b,j]*w1[u] + b1[u])
// block (128, 2): threadIdx.x = u, threadIdx.y = local batch. grid 128.
// c[] and f[b,:] staged through LDS in 128-wide tiles; inner loop consumes
// them as float4 (ds_load_b128) to cut DS issue count 4x.
// ---------------------------------------------------------------------------
__global__ void node_reduce_kernel(const float* __restrict__ f,
                                   const float* __restrict__ c,
                                   const float* __restrict__ w1,
                                   const float* __restrict__ b1,
                                   float* __restrict__ xmean) {
    __shared__ __align__(16) float sc[128];
    __shared__ __align__(16) float sf[2][128];
    int u  = threadIdx.x;
    int by = threadIdx.y;
    int b  = blockIdx.x * 2 + by;

    float w    = w1[u];
    float bias = b1[u];
    float acc  = 0.0f;

    const float4* scv = (const float4*)sc;
    const float4* sfv = (const float4*)sf[by];

    for (int j0 = 0; j0 < NN; j0 += 128) {
        __syncthreads();
        if (by == 0) sc[u] = c[j0 + u];
        sf[by][u] = f[(size_t)b * NN + j0 + u];
        __syncthreads();
        #pragma unroll 4
        for (int jj = 0; jj < 32; ++jj) {
            float4 fv = sfv[jj];
            float4 cv = scv[jj];
            float x0 = fmaf(fv.x, w, bias); x0 = x0 > 0.f ? x0 : 0.f;
            float x1 = fmaf(fv.y, w, bias); x1 = x1 > 0.f ? x1 : 0.f;
            float x2 = fmaf(fv.z, w, bias); x2 = x2 > 0.f ? x2 : 0.f;
            float x3 = fmaf(fv.w, w, bias); x3 = x3 > 0.f ? x3 : 0.f;
            acc = fmaf(cv.x, x0, acc);
            acc = fmaf(cv.y, x1, acc);
            acc = fmaf(cv.z, x2, acc);
            acc = fmaf(cv.w, x3, acc);
        }
    }
    xmean[b * UU + u] = acc;
}

// ---------------------------------------------------------------------------
// Kernel 3: out = relu(x_mean @ w2 + b2) via V_WMMA_F32_16X16X4_F32.
// M=256 (b), K=128 (u), N=256 (g). One 16x16 tile per wave32; 256 waves.
// A 16x4 layout (ISA 7.12.2): lane m = lane%16, K pair = (lane>=16 ? {2,3} : {0,1}).
// C/D layout: VGPR r -> M = r (lanes 0-15) / r+8 (lanes 16-31), N = lane%16.
// ---------------------------------------------------------------------------
__global__ void graph_dense_wmma_kernel(const float* __restrict__ xmean,
                                        const float* __restrict__ w2,
                                        const float* __restrict__ b2,
                                        float* __restrict__ out) {
    int wave  = (blockIdx.x * blockDim.x + threadIdx.x) >> 5;  // 0..255
    int lane  = threadIdx.x & 31;
    int tb    = (wave >> 4) * 16;   // batch-tile base
    int tg    = (wave & 15) * 16;   // graph-unit tile base
    int m     = lane & 15;
    int khalf = (lane >> 4) * 2;    // 0 for lanes 0-15, 2 for lanes 16-31

    v8f acc = {};
    #pragma unroll 4
    for (int k = 0; k < UU; k += 4) {
        const float* ap = xmean + (size_t)(tb + m) * UU + k + khalf;
        v2f a;
        a.x = ap[0];
        a.y = ap[1];
        const float* bp = w2 + (size_t)(k + khalf) * GG + tg + m;
        v2f bm;
        bm.x = bp[0];
        bm.y = bp[GG];
        acc = __builtin_amdgcn_wmma_f32_16x16x4_f32(
            /*neg_a=*/false, a, /*neg_b=*/false, bm,
            /*c_mod=*/(short)0, acc, /*reuse_a=*/false, /*reuse_b=*/false);
    }

    int col = tg + m;
    float bias = b2[col];
    int rbase = tb + ((lane >> 4) * 8);
    #pragma unroll
    for (int r = 0; r < 8; ++r) {
        float v = acc[r] + bias;
        out[(size_t)(rbase + r) * GG + col] = v > 0.0f ? v : 0.0f;
    }
}

// ---------------------------------------------------------------------------
// Launch. Inputs (setup_inputs order):
//   d_in[0] node_features (B*N f32), d_in[1] adj (N*N f32),
//   d_in[2] w1 (128 f32), d_in[3] b1 (128 f32),
//   d_in[4] w2 (128*256 f32), d_in[5] b2 (256 f32)
// ws layout (floats): c[0..2047], xmean[2048..2048+32767]
// (xmean region doubles as the colsum partial buffer — consumed before reuse)
// ---------------------------------------------------------------------------
extern "C" void kernel_launch(void* const* d_in, const int* in_sizes, int n_in,
                              void* d_out, int out_size, void* d_ws, size_t ws_size,
                              hipStream_t stream) {
    const float* f   = (const float*)d_in[0];
    const float* adj = (const float*)d_in[1];
    const float* w1  = (const float*)d_in[2];
    const float* b1  = (const float*)d_in[3];
    const float* w2  = (const float*)d_in[4];
    const float* b2  = (const float*)d_in[5];
    float* out = (float*)d_out;

    float* ws      = (float*)d_ws;
    float* c       = ws;              // NN floats
    float* xmean   = ws + NN;         // BB*UU floats
    float* partial = ws + NN;         // 16*(NN) floats, reused before xmean written

    colsum_partial_kernel<<<dim3(NN / 1024, 16), 256, 0, stream>>>(
        (const float4*)adj, (float4*)partial);
    colsum_reduce_kernel<<<dim3(NN / 1024), 256, 0, stream>>>(
        (const float4*)partial, (float4*)c);
    node_reduce_kernel<<<dim3(BB / 2), dim3(128, 2), 0, stream>>>(f, c, w1, b1, xmean);
    graph_dense_wmma_kernel<<<dim3(32), 256, 0, stream>>>(xmean, w2, b2, out);
}